// TopK_19756849562156
// MI455X (gfx1250) — compile-verified
//
#include <hip/hip_runtime.h>

// Sinkhorn soft-top-K, B=512 rows x N=4096, 2 anchors, 200 iterations.
// One 256-thread block per row; G kept in VGPRs (32 regs/thread);
// per-iteration block reduction of 2 scalars via V_WMMA_F32_16X16X4_F32
// (exact f32) + one barrier (double-buffered LDS slots).

#define BN    512
#define NN    4096
#define KK    256
#define EPSV  0.1f
#define ITERS 200
#define PADV  1e-16f

typedef __attribute__((ext_vector_type(2))) float v2f;
typedef __attribute__((ext_vector_type(8))) float v8f;

// ---------------- workspace init (harness poisons ws with 0xAA) --------------
__global__ void ws_init_kernel(int* ws) {
    if (blockIdx.x == 0 && threadIdx.x == 0) ws[0] = 0;
}

// ---------------- global Cmax = max over all (b,n) of max(s^2,(s-1)^2) ------
// Values are >= 0, so integer max on the float bit pattern == float max.
__global__ __launch_bounds__(256) void cmax_kernel(const float* __restrict__ s,
                                                   int* __restrict__ ws, int n) {
    int idx    = blockIdx.x * blockDim.x + threadIdx.x;
    int stride = gridDim.x * blockDim.x;
    float m = 0.0f;
    for (int i = idx; i < n; i += stride) {
        float x  = s[i];
        float x1 = x - 1.0f;
        m = fmaxf(m, fmaxf(x * x, x1 * x1));
    }
#pragma unroll
    for (int off = 16; off >= 1; off >>= 1)
        m = fmaxf(m, __shfl_xor(m, off, 32));
    if ((threadIdx.x & 31) == 0)
        atomicMax(ws, __float_as_int(m));
}

// ---------------- block-wide reduction of two f32 partials -------------------
// Intra-wave: V_WMMA_F32_16X16X4_F32 with B = ones (layout-robust): each D row
// equals s(m)+s(m+16), replicated across columns; summing the 8 D VGPRs per
// lane + one xor-16 shuffle yields the full wave32 sum in every lane.
// Cross-wave: one LDS exchange + one barrier (slot double-buffered by caller).
__device__ __forceinline__ void block_reduce2(float in0, float in1,
                                              float* red, int slotOfs,
                                              int wid, int lane,
                                              float& S0, float& S1) {
    v2f a0; a0[0] = in0;  a0[1] = 0.0f;
    v2f a1; a1[0] = in1;  a1[1] = 0.0f;
    v2f bo; bo[0] = 1.0f; bo[1] = 1.0f;
    v8f c0 = {};
    v8f c1 = {};
    c0 = __builtin_amdgcn_wmma_f32_16x16x4_f32(false, a0, false, bo, (short)0, c0, false, false);
    c1 = __builtin_amdgcn_wmma_f32_16x16x4_f32(false, a1, false, bo, (short)0, c1, false, false);
    float r0 = ((c0[0] + c0[1]) + (c0[2] + c0[3])) + ((c0[4] + c0[5]) + (c0[6] + c0[7]));
    float r1 = ((c1[0] + c1[1]) + (c1[2] + c1[3])) + ((c1[4] + c1[5]) + (c1[6] + c1[7]));
    r0 += __shfl_xor(r0, 16, 32);
    r1 += __shfl_xor(r1, 16, 32);
    if (lane == 0) {
        red[slotOfs + 2 * wid + 0] = r0;
        red[slotOfs + 2 * wid + 1] = r1;
    }
    __syncthreads();
    float t0 = 0.0f, t1 = 0.0f;
#pragma unroll
    for (int w = 0; w < 8; ++w) {
        t0 += red[slotOfs + 2 * w + 0];
        t1 += red[slotOfs + 2 * w + 1];
    }
    S0 = t0;
    S1 = t1;
}

// ---------------- main Sinkhorn kernel: one block per row --------------------
__global__ __launch_bounds__(256) void sinkhorn_kernel(const float* __restrict__ scores,
                                                       float* __restrict__ out,
                                                       const int* __restrict__ ws) {
    __shared__ float red[32];  // 2 slots x 8 waves x 2 scalars
    const int b    = blockIdx.x;
    const int t    = threadIdx.x;
    const int wid  = t >> 5;
    const int lane = t & 31;

    const float cmax = __int_as_float(ws[0]);
    const float coef = -1.0f / (EPSV * cmax);  // G = exp(coef * (s-a)^2)

    // Load 16 score values per thread (float4, fully coalesced), build G in VGPRs.
    // Register r = 4*j + c  <->  n = 4*t + 1024*j + c  (same mapping for stores).
    float G0[16], G1[16];
    const float* srow = scores + (size_t)b * NN;
#pragma unroll
    for (int j = 0; j < 4; ++j) {
        float4 sv = ((const float4*)srow)[t + j * 256];
        float xs[4] = {sv.x, sv.y, sv.z, sv.w};
#pragma unroll
        for (int c = 0; c < 4; ++c) {
            float x  = xs[c];
            float x1 = x - 1.0f;
            G0[4 * j + c] = __expf(coef * x * x);
            G1[4 * j + c] = __expf(coef * x1 * x1);
        }
    }

    const float mu  = 1.0f / (float)NN;
    const float nu0 = (float)KK / (float)NN;
    const float nu1 = (float)(NN - KK) / (float)NN;

    // Iteration 1: v = ones  =>  S_k = sum_n G[k,n]
    float s0 = 0.0f, s1 = 0.0f;
#pragma unroll
    for (int j = 0; j < 16; ++j) { s0 += G0[j]; s1 += G1[j]; }
    float S0, S1;
    block_reduce2(s0, s1, red, 0, wid, lane, S0, S1);
    float u0 = nu0 * __builtin_amdgcn_rcpf(S0 + PADV);
    float u1 = nu1 * __builtin_amdgcn_rcpf(S1 + PADV);

    // Iterations 2..200: v eliminated, only u (2 scalars) is live across threads.
    for (int it = 1; it < ITERS; ++it) {
        s0 = 0.0f; s1 = 0.0f;
#pragma unroll
        for (int j = 0; j < 16; ++j) {
            float d  = fmaf(G0[j], u0, fmaf(G1[j], u1, PADV));
            float tv = mu * __builtin_amdgcn_rcpf(d);
            s0 = fmaf(G0[j], tv, s0);
            s1 = fmaf(G1[j], tv, s1);
        }
        block_reduce2(s0, s1, red, (it & 1) * 16, wid, lane, S0, S1);
        u0 = nu0 * __builtin_amdgcn_rcpf(S0 + PADV);
        u1 = nu1 * __builtin_amdgcn_rcpf(S1 + PADV);
    }

    // Final v from final u, then P = u * G * v ; coalesced float4 stores.
    float* out0 = out + (size_t)b * 2 * NN;
    float* out1 = out0 + NN;
#pragma unroll
    for (int j = 0; j < 4; ++j) {
        float4 p0, p1;
        float* pp0 = (float*)&p0;
        float* pp1 = (float*)&p1;
#pragma unroll
        for (int c = 0; c < 4; ++c) {
            int r    = 4 * j + c;
            float d  = fmaf(G0[r], u0, fmaf(G1[r], u1, PADV));
            float v  = mu * __builtin_amdgcn_rcpf(d);
            pp0[c] = u0 * G0[r] * v;
            pp1[c] = u1 * G1[r] * v;
        }
        ((float4*)out0)[t + j * 256] = p0;
        ((float4*)out1)[t + j * 256] = p1;
    }
}

// ---------------- entry point ------------------------------------------------
extern "C" void kernel_launch(void* const* d_in, const int* in_sizes, int n_in,
                              void* d_out, int out_size, void* d_ws, size_t ws_size,
                              hipStream_t stream) {
    const float* scores = (const float*)d_in[0];
    float* out = (float*)d_out;
    int* ws = (int*)d_ws;

    ws_init_kernel<<<1, 1, 0, stream>>>(ws);
    cmax_kernel<<<1024, 256, 0, stream>>>(scores, ws, BN * NN);
    sinkhorn_kernel<<<BN, 256, 0, stream>>>(scores, out, ws);
}